// TwoWayGCNConv_28338194219470
// MI455X (gfx1250) — compile-verified
//
#include <hip/hip_runtime.h>
#include <stdint.h>

#define N_NODES 20000
#define N_EDGES 50000
#define NUM_RELS 32
#define C 64              // IN_C == OUT_C == 64
#define WSQ (C * C)       // 4096 floats per relation weight

typedef __attribute__((ext_vector_type(2))) float v2f;
typedef __attribute__((ext_vector_type(8))) float v8f;

// D = A(16x4) * B(4x16) + C(16x16), all f32, wave32.
__device__ __forceinline__ v8f wmma4(v2f a, v2f b, v8f c) {
    return __builtin_amdgcn_wmma_f32_16x16x4_f32(
        /*neg_a=*/false, a, /*neg_b=*/false, b,
        /*c_mod=*/(short)0, c, /*reuse_a=*/false, /*reuse_b=*/false);
}

// ---------------------------------------------------------------- utilities
__global__ void zero_ws_kernel(unsigned int* p, int n) {
    int i = blockIdx.x * blockDim.x + threadIdx.x;
    if (i < n) p[i] = 0u;
}

__global__ void count_kernel(const int* __restrict__ src, const int* __restrict__ dst,
                             const int* __restrict__ et,
                             unsigned int* __restrict__ cs, unsigned int* __restrict__ cd,
                             unsigned int* __restrict__ rc) {
    int e = blockIdx.x * blockDim.x + threadIdx.x;
    if (e < N_EDGES) {
        atomicAdd(&cs[src[e]], 1u);
        atomicAdd(&cd[dst[e]], 1u);
        atomicAdd(&rc[et[e]], 1u);
    }
}

// one block per (table, relation): inv_norm = 1/(||W_rel||_F + 0.01)
__global__ __launch_bounds__(256) void norm_kernel(const float* __restrict__ wf,
                                                   const float* __restrict__ wb,
                                                   float* __restrict__ inf,
                                                   float* __restrict__ inb) {
    __shared__ float red[256];
    int b = blockIdx.x;                 // 0..63
    int r = b & (NUM_RELS - 1);
    const float* p = (b < NUM_RELS ? wf : wb) + (size_t)r * WSQ;
    float s = 0.f;
    for (int i = threadIdx.x; i < WSQ; i += 256) { float v = p[i]; s += v * v; }
    red[threadIdx.x] = s;
    __syncthreads();
    for (int off = 128; off > 0; off >>= 1) {
        if (threadIdx.x < off) red[threadIdx.x] += red[threadIdx.x + off];
        __syncthreads();
    }
    if (threadIdx.x == 0) {
        float invn = 1.f / (sqrtf(red[0]) + 0.01f);
        if (b < NUM_RELS) inf[r] = invn; else inb[r] = invn;
    }
}

__global__ void scan_kernel(const unsigned int* __restrict__ rc,
                            unsigned int* __restrict__ off,
                            unsigned int* __restrict__ cur) {
    if (threadIdx.x == 0) {
        unsigned int a = 0;
        for (int r = 0; r < NUM_RELS; ++r) { off[r] = a; a += rc[r]; }
        off[NUM_RELS] = a;
    }
    if (threadIdx.x < NUM_RELS) cur[threadIdx.x] = 0u;
}

__global__ void bucket_kernel(const int* __restrict__ et,
                              const unsigned int* __restrict__ off,
                              unsigned int* __restrict__ cur,
                              unsigned int* __restrict__ order) {
    int e = blockIdx.x * blockDim.x + threadIdx.x;
    if (e < N_EDGES) {
        int t = et[e];
        unsigned int p = atomicAdd(&cur[t], 1u);
        order[off[t] + p] = (unsigned int)e;
    }
}

// ------------------------------------------------- out = x @ lin_w.T + lin_b
// 8 waves/block, one 16-row tile per wave. B[k][n] = lin_w[n][k] -> LDS [n][k],
// padded stride 65 for bank-conflict-free access.
__global__ __launch_bounds__(256) void lin_wmma_kernel(const float* __restrict__ x,
                                                       const float* __restrict__ lw,
                                                       const float* __restrict__ lb,
                                                       float* __restrict__ out) {
    __shared__ float Wl[C * (C + 1)];
    __shared__ float Bl[C];
    int tid = threadIdx.x;
    for (int j = tid; j < WSQ; j += 256) {       // lw is [n][k] row-major already
        int n = j >> 6, k = j & 63;
        Wl[n * 65 + k] = lw[j];
    }
    if (tid < C) Bl[tid] = lb[tid];
    __syncthreads();

    int wave = tid >> 5, lane = tid & 31, m = lane & 15, hi = lane >> 4;
    int tile = blockIdx.x * 8 + wave;
    if (tile >= N_NODES / 16) return;            // wave-uniform exit

    const float* xr = x + (size_t)(tile * 16 + m) * C + 2 * hi;
    v8f a0 = {}, a1 = {}, a2 = {}, a3 = {};
#pragma unroll
    for (int f = 0; f < 16; ++f) {
        v2f a = *(const v2f*)(xr + 4 * f);
        int k = 4 * f + 2 * hi;
        v2f b0 = { Wl[(m) * 65 + k],      Wl[(m) * 65 + k + 1] };
        v2f b1 = { Wl[(16 + m) * 65 + k], Wl[(16 + m) * 65 + k + 1] };
        v2f b2 = { Wl[(32 + m) * 65 + k], Wl[(32 + m) * 65 + k + 1] };
        v2f b3 = { Wl[(48 + m) * 65 + k], Wl[(48 + m) * 65 + k + 1] };
        a0 = wmma4(a, b0, a0);
        a1 = wmma4(a, b1, a1);
        a2 = wmma4(a, b2, a2);
        a3 = wmma4(a, b3, a3);
    }
#pragma unroll
    for (int i = 0; i < 8; ++i) {
        size_t ro = (size_t)(tile * 16 + i + 8 * hi) * C;
        out[ro + m]      = a0[i] + Bl[m];
        out[ro + 16 + m] = a1[i] + Bl[16 + m];
        out[ro + 32 + m] = a2[i] + Bl[32 + m];
        out[ro + 48 + m] = a3[i] + Bl[48 + m];
    }
}

// ----------------------------------------------- per-relation batched conv
// grid (NUM_RELS, SPLITS, 2): z = direction. Each block caches W[rel] in LDS
// (transposed [n][k], stride 65); each wave processes 16-edge tiles via WMMA
// and atomically scatter-adds scaled rows into out[dst].
__global__ __launch_bounds__(256) void conv_wmma_kernel(
    const float* __restrict__ x,
    const int* __restrict__ ei0, const int* __restrict__ ei1,
    const float* __restrict__ wf, const float* __restrict__ wb,
    const unsigned int* __restrict__ counts_src, const unsigned int* __restrict__ counts_dst,
    const unsigned int* __restrict__ rel_off, const unsigned int* __restrict__ order,
    const float* __restrict__ invn_f, const float* __restrict__ invn_b,
    float* __restrict__ out) {
    __shared__ float Wl[C * (C + 1)];
    int dir = blockIdx.z;
    int r = blockIdx.x;
    const float* Wt = (dir == 0 ? wf : wb) + (size_t)r * WSQ;
    const int* srcIdx = dir == 0 ? ei0 : ei1;
    const int* dstIdx = dir == 0 ? ei1 : ei0;
    const unsigned int* cnt = dir == 0 ? counts_src : counts_dst;
    float invn = (dir == 0 ? invn_f : invn_b)[r];

    int tid = threadIdx.x;
    for (int j = tid; j < WSQ; j += 256) {       // Wt is [k][n]; store [n][k]
        int k = j >> 6, n = j & 63;
        Wl[n * 65 + k] = Wt[j];
    }
    __syncthreads();

    unsigned int begin = rel_off[r], end = rel_off[r + 1];
    unsigned int nE = end - begin;
    unsigned int nT = (nE + 15u) >> 4;

    int wave = tid >> 5, lane = tid & 31, m = lane & 15, hi = lane >> 4;
    unsigned int g = blockIdx.y * 8 + wave;
    unsigned int stride = gridDim.y * 8;

    for (unsigned int t = g; t < nT; t += stride) {
        unsigned int idx = t * 16 + (unsigned int)m;
        bool valid = idx < nE;
        unsigned int eo = valid ? begin + idx : begin;   // pad rows reuse a valid edge
        int eid = (int)order[eo];
        int s = srcIdx[eid];
        int d = dstIdx[eid];
        float sc = valid ? invn / (float)(cnt[s] + 1u) : 0.f;  // pad rows add 0

        const float* xr = x + (size_t)s * C + 2 * hi;
        v8f a0 = {}, a1 = {}, a2 = {}, a3 = {};
#pragma unroll
        for (int f = 0; f < 16; ++f) {
            v2f a = *(const v2f*)(xr + 4 * f);
            int k = 4 * f + 2 * hi;
            v2f b0 = { Wl[(m) * 65 + k],      Wl[(m) * 65 + k + 1] };
            v2f b1 = { Wl[(16 + m) * 65 + k], Wl[(16 + m) * 65 + k + 1] };
            v2f b2 = { Wl[(32 + m) * 65 + k], Wl[(32 + m) * 65 + k + 1] };
            v2f b3 = { Wl[(48 + m) * 65 + k], Wl[(48 + m) * 65 + k + 1] };
            a0 = wmma4(a, b0, a0);
            a1 = wmma4(a, b1, a1);
            a2 = wmma4(a, b2, a2);
            a3 = wmma4(a, b3, a3);
        }
#pragma unroll
        for (int i = 0; i < 8; ++i) {
            int rsel = i + 8 * hi;                      // D row held by this VGPR
            int drow = __shfl(d, rsel);
            float scr = __shfl(sc, rsel);
            size_t base = (size_t)drow * C;
            atomicAdd(&out[base + m],      a0[i] * scr);
            atomicAdd(&out[base + 16 + m], a1[i] * scr);
            atomicAdd(&out[base + 32 + m], a2[i] * scr);
            atomicAdd(&out[base + 48 + m], a3[i] * scr);
        }
    }
}

// --------------------------------------------------------------------------
extern "C" void kernel_launch(void* const* d_in, const int* in_sizes, int n_in,
                              void* d_out, int out_size, void* d_ws, size_t ws_size,
                              hipStream_t stream) {
    const float* x  = (const float*)d_in[0];
    const int*   ei = (const int*)d_in[1];           // [2, N_EDGES] flat
    const int*   et = (const int*)d_in[2];
    const float* wf = (const float*)d_in[3];
    const float* wb = (const float*)d_in[4];
    const float* lw = (const float*)d_in[5];
    const float* lb = (const float*)d_in[6];
    float* out = (float*)d_out;
    const int* ei0 = ei;                              // src row
    const int* ei1 = ei + N_EDGES;                    // dst row

    // workspace carve-up (~361 KB)
    unsigned int* counts_src = (unsigned int*)d_ws;                   // N_NODES
    unsigned int* counts_dst = counts_src + N_NODES;                  // N_NODES
    unsigned int* rel_count  = counts_dst + N_NODES;                  // 32
    unsigned int* rel_off    = rel_count + NUM_RELS;                  // 33
    unsigned int* rel_cur    = rel_off + NUM_RELS + 1;                // 32
    unsigned int* order      = rel_cur + NUM_RELS;                    // N_EDGES
    float* invn_f = (float*)(order + N_EDGES);                        // 32
    float* invn_b = invn_f + NUM_RELS;                                // 32

    int zero_words = 2 * N_NODES + NUM_RELS + (NUM_RELS + 1) + NUM_RELS;
    zero_ws_kernel<<<(zero_words + 255) / 256, 256, 0, stream>>>(counts_src, zero_words);

    count_kernel<<<(N_EDGES + 255) / 256, 256, 0, stream>>>(
        ei0, ei1, et, counts_src, counts_dst, rel_count);

    norm_kernel<<<2 * NUM_RELS, 256, 0, stream>>>(wf, wb, invn_f, invn_b);

    scan_kernel<<<1, 32, 0, stream>>>(rel_count, rel_off, rel_cur);

    bucket_kernel<<<(N_EDGES + 255) / 256, 256, 0, stream>>>(et, rel_off, rel_cur, order);

    // initializes every element of d_out (x @ lin_w.T + lin_b)
    lin_wmma_kernel<<<(N_NODES / 16 + 7) / 8, 256, 0, stream>>>(x, lw, lb, out);

    // fwd (z=0) and bwd (z=1) convs accumulate on top
    dim3 cgrid(NUM_RELS, 4, 2);
    conv_wmma_kernel<<<cgrid, 256, 0, stream>>>(
        x, ei0, ei1, wf, wb, counts_src, counts_dst, rel_off, order,
        invn_f, invn_b, out);
}